// RelativePositionalEmbedding2DKey_84688165143032
// MI455X (gfx1250) — compile-verified
//
#include <hip/hip_runtime.h>

typedef __attribute__((ext_vector_type(2))) float v2f;
typedef __attribute__((ext_vector_type(4))) float fvec4;
typedef __attribute__((ext_vector_type(8))) float v8f;

#define T_   4
#define H_   16
#define W_   16
#define C_   64
#define BN_  64        // BATCH*HEADS
#define KTOK 1024      // T*H*W

// ---------------------------------------------------------------------------
// Stage 1: bias GEMMs via V_WMMA_F32_16X16X4_F32.
// One wave (32 lanes) per 16x16 output tile, K=64 -> 16 chained WMMAs.
//   kind 0 (H): fixed (kh,kt): A rows = 16 kw tokens (contig),  B = Rh[kh]  (N=qh)
//   kind 1 (W): fixed (kw,kt): A rows = 16 kh tokens (stride16),B = Rw[kw]  (N=qw)
//   kind 2 (T): fixed kt, 16-token tile: B = Rt[kt] (N=qt, cols qt>=4 zeroed
//               via branch-free 0/1 mask multiply — keeps EXEC untouched)
// 4096 jobs per kind, 8 waves per 256-thread block.
// ---------------------------------------------------------------------------
__global__ __launch_bounds__(256)
void rpe_bias_wmma_kernel(const float* __restrict__ key,
                          const float* __restrict__ embH,
                          const float* __restrict__ embW,
                          const float* __restrict__ embT,
                          float* __restrict__ relH,
                          float* __restrict__ relW,
                          float* __restrict__ relT)
{
    const int lane = threadIdx.x & 31;
    const int wave = threadIdx.x >> 5;
    const int job  = blockIdx.x * 8 + wave;      // 0 .. 12287

    const int n    = lane & 15;                  // B column / D column (q index)
    const int half = lane >> 4;                  // 0/1 -> K sub-pair & D row half
    const int kind = job >> 12;                  // 0=H, 1=W, 2=T
    const int r    = job & 4095;
    const int bn   = r >> 6;                     // batch*head
    const int sub  = r & 63;

    const float* aBase;
    const float* bRow;
    float*       dBase;
    int   aStrideTok;      // floats between consecutive A rows (tokens)
    int   dStrideM;        // floats between consecutive D rows in output
    float bMask = 1.0f;    // per-lane 0/1 mask for B (branch-free zeroing)

    if (kind == 0) {
        const int kh = sub >> 2, kt = sub & 3;
        aBase = key + (size_t)(bn * KTOK + kt * 256 + kh * 16) * C_;
        aStrideTok = C_;                               // token = kw, contiguous
        bRow  = embH + (size_t)(kh - n + (H_ - 1)) * C_;
        dBase = relH + ((size_t)bn * 16 + n) * KTOK + (kt * 256 + kh * 16);
        dStrideM = 1;
    } else if (kind == 1) {
        const int kw = sub >> 2, kt = sub & 3;
        aBase = key + (size_t)(bn * KTOK + kt * 256 + kw) * C_;
        aStrideTok = 16 * C_;                          // token = kh, stride 16
        bRow  = embW + (size_t)(kw - n + (W_ - 1)) * C_;
        dBase = relW + ((size_t)bn * 16 + n) * KTOK + (kt * 256 + kw);
        dStrideM = 16;
    } else {
        const int  kt = sub >> 4, tile = sub & 15;
        const bool bValid = (n < T_);
        aBase = key + (size_t)(bn * KTOK + kt * 256 + tile * 16) * C_;
        aStrideTok = C_;
        bMask = bValid ? 1.0f : 0.0f;
        bRow  = embT + (size_t)(bValid ? (kt - n + (T_ - 1)) : 0) * C_; // clamped, safe
        dBase = relT + ((size_t)bn * T_ + n) * KTOK + (kt * 256 + tile * 16);
        dStrideM = 1;
    }

    // Per-lane pointers per ISA A/B VGPR layouts (M/N = lane&15, K = j*4 + half*2 + {0,1})
    const float* aPtr = aBase + (size_t)(lane & 15) * aStrideTok + half * 2;
    const float* bPtr = bRow + half * 2;

    v8f acc = {0.f, 0.f, 0.f, 0.f, 0.f, 0.f, 0.f, 0.f};

#pragma unroll
    for (int j = 0; j < 16; ++j) {                    // K = 64 = 16 * 4
        v2f a = *(const v2f*)(aPtr + j * 4);
        v2f b = *(const v2f*)(bPtr + j * 4);          // unconditional load
        b.x *= bMask;                                 // zero invalid qt columns
        b.y *= bMask;
        // (neg_a, A, neg_b, B, c_mod, C, reuse_a, reuse_b)
        acc = __builtin_amdgcn_wmma_f32_16x16x4_f32(false, a, false, b,
                                                    (short)0, acc, false, false);
    }

    if (kind == 2 && n >= T_) return;                 // unused qt columns

    // D layout: VGPR v -> row M = v + 8*half, column N = lane&15 (already in dBase)
    float* p = dBase + (size_t)(half * 8) * dStrideM;
    if (dStrideM == 1) {
        fvec4 lo = {acc[0], acc[1], acc[2], acc[3]};
        fvec4 hi = {acc[4], acc[5], acc[6], acc[7]};
        *(fvec4*)(p)     = lo;                        // 32B-aligned (offset mult of 8 floats)
        *(fvec4*)(p + 4) = hi;
    } else {
#pragma unroll
        for (int v = 0; v < 8; ++v) p[(size_t)v * dStrideM] = acc[v];
    }
}

// ---------------------------------------------------------------------------
// Stage 2: out[bn][q][k] = scores + relH[bn][qh][k] + relW[bn][qw][k] + relT[bn][qt][k]
// Pure HBM stream (512 MB): float4 per thread, non-temporal on the 256 MB
// streams so the 9 MB of bias rows (reused 64x, relT rows 256x) stay resident in L2.
// ---------------------------------------------------------------------------
__global__ __launch_bounds__(256)
void rpe_add_kernel(const float* __restrict__ scores,
                    const float* __restrict__ relH,
                    const float* __restrict__ relW,
                    const float* __restrict__ relT,
                    float* __restrict__ out)
{
    const size_t idx = (size_t)blockIdx.x * 256 + threadIdx.x;  // one float4 each
    const int    k4  = (int)(idx & 255);                        // 1024/4 chunks
    const size_t row = idx >> 8;                                // bn*1024 + q
    const int    q   = (int)(row & 1023);
    const int    bn  = (int)(row >> 10);
    const int    qw  = q & 15;
    const int    qh  = (q >> 4) & 15;
    const int    qt  = q >> 8;

    const fvec4* s4 = (const fvec4*)(scores + row * KTOK) + k4;
    const fvec4* h4 = (const fvec4*)(relH + ((size_t)bn * 16 + qh) * KTOK) + k4;
    const fvec4* w4 = (const fvec4*)(relW + ((size_t)bn * 16 + qw) * KTOK) + k4;
    const fvec4* t4 = (const fvec4*)(relT + ((size_t)bn * T_ + qt) * KTOK) + k4;

    fvec4 s = __builtin_nontemporal_load(s4);
    fvec4 o = s + *h4 + *w4 + *t4;
    __builtin_nontemporal_store(o, (fvec4*)(out + row * KTOK) + k4);
}

extern "C" void kernel_launch(void* const* d_in, const int* in_sizes, int n_in,
                              void* d_out, int out_size, void* d_ws, size_t ws_size,
                              hipStream_t stream) {
    const float* key    = (const float*)d_in[0];  // (8,8,1024,64)
    const float* scores = (const float*)d_in[1];  // (8,8,1024,1024)
    const float* embH   = (const float*)d_in[2];  // (31,64)
    const float* embW   = (const float*)d_in[3];  // (31,64)
    const float* embT   = (const float*)d_in[4];  // (7,64)
    float*       out    = (float*)d_out;

    float* ws   = (float*)d_ws;
    float* relH = ws;                              // 64*16*1024 floats (4 MB)
    float* relW = ws + (size_t)BN_ * 16 * KTOK;    // 4 MB
    float* relT = ws + (size_t)2 * BN_ * 16 * KTOK;// 1 MB   (total 9 MB)

    // 12288 wave-jobs, 8 waves per block
    rpe_bias_wmma_kernel<<<1536, 256, 0, stream>>>(key, embH, embW, embT,
                                                   relH, relW, relT);

    // 64M floats / 4 per thread / 256 per block
    rpe_add_kernel<<<65536, 256, 0, stream>>>(scores, relH, relW, relT, out);
}